// Model_50182397886770
// MI455X (gfx1250) — compile-verified
//
#include <hip/hip_runtime.h>

// ---------------------------------------------------------------------------
// Wavetable oscillator: linear-upsampled rate -> cumsum (phase) -> fractional
// wraparound lookup. Hierarchical scan; the 256-wide block scan is done with
// V_WMMA_F32_16X16X4_F32 (f32 "tensor-core scan": P = X*U ; D = L'*T + P).
// ---------------------------------------------------------------------------

typedef __attribute__((ext_vector_type(2))) float v2f;
typedef __attribute__((ext_vector_type(8))) float v8f;

#define N_SAMPLES   16777216        // 2^24
#define N_FRAMES    2048
#define WT_SIZE     4096
#define CHUNK       4096            // samples per block
#define THREADS     256
#define PER_THREAD  16              // CHUNK / THREADS
#define N_BLOCKS    (N_SAMPLES / CHUNK)   // 4096

// rate[n] = linear_upsample_1d(p, N_SAMPLES)[n]; scale = 2048/2^24 = 2^-13 exact
__device__ __forceinline__ float rate_at(const float* __restrict__ s_p, int n) {
    float pos = ((float)n + 0.5f) * (1.0f / 8192.0f) - 0.5f;
    pos = fminf(fmaxf(pos, 0.0f), 2047.0f);
    float fl = floorf(pos);
    int i0 = (int)fl;
    int i1 = (i0 + 1 < 2047) ? (i0 + 1) : 2047;
    float fr = pos - fl;
    float p0 = s_p[i0];
    return p0 + fr * (s_p[i1] - p0);
}

// Inclusive scan of 256 floats (one per thread) via 8x V_WMMA_F32_16X16X4_F32.
// All 256 threads must call (wave 0 does the matrix work; EXEC all-ones there).
// In: s_sums[tid].  Out: s_scan[tid] = sum(s_sums[0..tid]).
__device__ __forceinline__ void block_scan256_wmma(const float* s_sums,
                                                   float* s_row,
                                                   float* s_scan) {
    const int tid  = threadIdx.x;
    __syncthreads();                 // all partials visible before wave 0 reads
    v8f acc = {};
    const int n    = tid & 15;       // column index; also A-operand row (M)
    const int half = tid >> 4;       // lane-half -> K pair select
    if (tid < 32) {
        // Pass 1: P = X * U  with U[k][c] = (k <= c)  -> row-wise inclusive scan
#pragma unroll
        for (int kk = 0; kk < 4; ++kk) {
            const int k0 = 4 * kk + 2 * half;
            v2f a, b;
            a.x = s_sums[n * 16 + k0];          // A: X[m=n][k0]
            a.y = s_sums[n * 16 + k0 + 1];
            b.x = (k0     <= n) ? 1.0f : 0.0f;  // B: U[k0][n]
            b.y = (k0 + 1 <= n) ? 1.0f : 0.0f;
            acc = __builtin_amdgcn_wmma_f32_16x16x4_f32(
                false, a, false, b, (short)0, acc, false, false);
        }
        if (n == 15) {                           // P[m][15] == row totals
#pragma unroll
            for (int v = 0; v < 8; ++v) s_row[v + 8 * half] = acc[v];
        }
    }
    __syncthreads();
    if (tid < 32) {
        // Pass 2: D = L' * T + P  (L'[m][k] = k<m ; T[k][*] = rowsum(k); C = P)
#pragma unroll
        for (int kk = 0; kk < 4; ++kk) {
            const int k0 = 4 * kk + 2 * half;
            v2f a, b;
            a.x = (k0     < n) ? 1.0f : 0.0f;   // A: L'[m=n][k0]
            a.y = (k0 + 1 < n) ? 1.0f : 0.0f;
            b.x = s_row[k0];                    // B: T[k0][n] (broadcast)
            b.y = s_row[k0 + 1];
            acc = __builtin_amdgcn_wmma_f32_16x16x4_f32(
                false, a, false, b, (short)0, acc, false, false);
        }
        // D layout: VGPR v, lanes 0-15 -> M=v ; lanes 16-31 -> M=v+8 ; N=lane%16
#pragma unroll
        for (int v = 0; v < 8; ++v)
            s_scan[(v + 8 * half) * 16 + n] = acc[v];
    }
    __syncthreads();
}

// ---------------------------------------------------------------------------
// Kernel A: per-chunk (4096-sample) rate totals.
// ---------------------------------------------------------------------------
__global__ __launch_bounds__(THREADS)
void k_chunk_sums(const float* __restrict__ p, float* __restrict__ chunk_sums) {
    __shared__ float s_p[N_FRAMES];
    __shared__ float s_sums[THREADS];
    __shared__ float s_row[16];
    __shared__ float s_scan[THREADS];
    const int tid = threadIdx.x;
#pragma unroll
    for (int j = 0; j < N_FRAMES / THREADS; ++j)
        s_p[tid + THREADS * j] = p[tid + THREADS * j];
    __syncthreads();

    const int n0 = blockIdx.x * CHUNK + tid * PER_THREAD;
    float s = 0.0f;
#pragma unroll
    for (int j = 0; j < PER_THREAD; ++j) s += rate_at(s_p, n0 + j);
    s_sums[tid] = s;

    block_scan256_wmma(s_sums, s_row, s_scan);
    if (tid == 0) chunk_sums[blockIdx.x] = s_scan[THREADS - 1];
}

// ---------------------------------------------------------------------------
// Kernel B: scan the 4096 chunk sums -> exclusive per-chunk phase bases.
// ---------------------------------------------------------------------------
__global__ __launch_bounds__(THREADS)
void k_base_scan(const float* __restrict__ chunk_sums, float* __restrict__ bases) {
    __shared__ float s_sums[THREADS];
    __shared__ float s_row[16];
    __shared__ float s_scan[THREADS];
    const int tid = threadIdx.x;
    float v[PER_THREAD];
    float s = 0.0f;
#pragma unroll
    for (int j = 0; j < PER_THREAD; ++j) {
        v[j] = chunk_sums[tid * PER_THREAD + j];
        s += v[j];
    }
    s_sums[tid] = s;

    block_scan256_wmma(s_sums, s_row, s_scan);

    float run = (tid == 0) ? 0.0f : s_scan[tid - 1];
#pragma unroll
    for (int j = 0; j < PER_THREAD; ++j) {
        bases[tid * PER_THREAD + j] = run;
        run += v[j];
    }
}

// ---------------------------------------------------------------------------
// Kernel C: rates -> intra-block WMMA scan -> phase -> LDS wavetable lookup,
// staged in LDS and written with fully coalesced float4 stores.
// ---------------------------------------------------------------------------
__global__ __launch_bounds__(THREADS)
void k_osc(const float* __restrict__ p, const float* __restrict__ wt,
           const float* __restrict__ bases, float* __restrict__ out) {
    __shared__ float s_p[N_FRAMES];
    __shared__ float s_wt[WT_SIZE];
    __shared__ float s_sums[THREADS];
    __shared__ float s_row[16];
    __shared__ float s_scan[THREADS];
    __shared__ __align__(16) float s_out[CHUNK];
    const int tid = threadIdx.x;
#pragma unroll
    for (int j = 0; j < N_FRAMES / THREADS; ++j)
        s_p[tid + THREADS * j] = p[tid + THREADS * j];
#pragma unroll
    for (int j = 0; j < WT_SIZE / THREADS; ++j)
        s_wt[tid + THREADS * j] = wt[tid + THREADS * j];
    __syncthreads();

    const int n0 = blockIdx.x * CHUNK + tid * PER_THREAD;
    float r[PER_THREAD];
    float s = 0.0f;
#pragma unroll
    for (int j = 0; j < PER_THREAD; ++j) {
        r[j] = rate_at(s_p, n0 + j);
        s += r[j];
    }
    s_sums[tid] = s;

    block_scan256_wmma(s_sums, s_row, s_scan);

    float phase = bases[blockIdx.x] + ((tid == 0) ? 0.0f : s_scan[tid - 1]);
#pragma unroll
    for (int j = 0; j < PER_THREAD; ++j) {
        phase += r[j];
        // floor-mod wrap into [0, 4096); & 4095 also fixes +/- rounding spill
        float q   = floorf(phase * (1.0f / 4096.0f));
        float idx = phase - 4096.0f * q;
        float lo  = floorf(idx);
        float fr  = idx - lo;
        int a0 = ((int)lo) & (WT_SIZE - 1);
        int a1 = (a0 + 1) & (WT_SIZE - 1);
        float w0 = s_wt[a0];
        s_out[tid * PER_THREAD + j] = w0 + fr * (s_wt[a1] - w0);
    }
    __syncthreads();

    float4* o4 = (float4*)(out + (size_t)blockIdx.x * CHUNK);
    const float4* l4 = (const float4*)s_out;
#pragma unroll
    for (int it = 0; it < CHUNK / 4 / THREADS; ++it)
        o4[tid + THREADS * it] = l4[tid + THREADS * it];
}

// ---------------------------------------------------------------------------
extern "C" void kernel_launch(void* const* d_in, const int* in_sizes, int n_in,
                              void* d_out, int out_size, void* d_ws, size_t ws_size,
                              hipStream_t stream) {
    (void)in_sizes; (void)n_in; (void)out_size; (void)ws_size;
    const float* p  = (const float*)d_in[1];   // d_in[0] = x (unused by fwd)
    const float* wt = (const float*)d_in[2];
    float* out = (float*)d_out;

    float* chunk_sums = (float*)d_ws;          // N_BLOCKS floats
    float* bases      = chunk_sums + N_BLOCKS; // N_BLOCKS floats (32 KB total)

    k_chunk_sums<<<N_BLOCKS, THREADS, 0, stream>>>(p, chunk_sums);
    k_base_scan<<<1, THREADS, 0, stream>>>(chunk_sums, bases);
    k_osc<<<N_BLOCKS, THREADS, 0, stream>>>(p, wt, bases, out);
}